// TileClassificationHead_25546465476636
// MI455X (gfx1250) — compile-verified
//
#include <hip/hip_runtime.h>

typedef __attribute__((ext_vector_type(2))) float v2f;
typedef __attribute__((ext_vector_type(8))) float v8f;

#define DIM      1536
#define NTILES   50000
#define NBLK     (NTILES / 16)   // 3125 sixteen-tile slabs
#define WAVES_PER_WG 8

// Kernel 1: per-wave 16-tile slab GEMV via V_WMMA_F32_16X16X4_F32.
//   A = fused weights [16 x 4] (rows 0..2 = Wc0, Wc1, Wa; rest zero)
//   B = h-slab        [4 x 16] (columns = tiles)
//   D = [16 x 16] f32, rows 0..2 = c0 / c1 / w per tile.
__global__ __launch_bounds__(256) void tile_head_wmma(
    const float* __restrict__ img,   // [NTILES, DIM]
    const float* __restrict__ Wc,    // [2, DIM]
    const float* __restrict__ bc,    // [2]
    const float* __restrict__ Wa,    // [1, DIM]
    const float* __restrict__ ba,    // [1]
    float* __restrict__ out,         // [1 + NTILES*2]
    float* __restrict__ partials)    // [NBLK, 2] (num, den)
{
    const int lane = threadIdx.x & 31;
    const int wave = threadIdx.x >> 5;
    const int blk  = blockIdx.x * WAVES_PER_WG + wave;   // uniform per wave
    if (blk >= NBLK) return;                             // whole-wave exit

    const int row   = lane & 15;        // A: M-row of Wf; B: tile column
    const int khalf = (lane >> 4) * 2;  // half-wave K offset {0,2}

    // A source row: rows 0..2 real, others read Wa (valid memory) then mask.
    const float* wrow = (row == 0) ? Wc : (row == 1) ? (Wc + DIM) : Wa;
    const float  wmask = (row < 3) ? 1.0f : 0.0f;

    const float* hrow = img + (size_t)(blk * 16 + row) * DIM;

    v8f acc0 = {};
    v8f acc1 = {};
#pragma unroll 2
    for (int k = 0; k < DIM; k += 8) {
        {
            const int kk = k + khalf;
            float2 av = *(const float2*)(wrow + kk);
            float2 bv = *(const float2*)(hrow + kk);
            v2f a; a.x = av.x * wmask; a.y = av.y * wmask;
            v2f b; b.x = bv.x;         b.y = bv.y;
            acc0 = __builtin_amdgcn_wmma_f32_16x16x4_f32(
                false, a, false, b, (short)0, acc0, false, false);
        }
        {
            const int kk = k + 4 + khalf;
            float2 av = *(const float2*)(wrow + kk);
            float2 bv = *(const float2*)(hrow + kk);
            v2f a; a.x = av.x * wmask; a.y = av.y * wmask;
            v2f b; b.x = bv.x;         b.y = bv.y;
            acc1 = __builtin_amdgcn_wmma_f32_16x16x4_f32(
                false, a, false, b, (short)0, acc1, false, false);
        }
    }
    v8f acc = acc0 + acc1;

    const float bc0 = bc[0];
    const float bc1 = bc[1];
    const float ba0 = ba[0];

    float pn = 0.0f, pd = 0.0f;
    if (lane < 16) {
        const float c0 = acc[0] + bc0;
        const float c1 = acc[1] + bc1;
        const float w  = acc[2] + ba0;
        const int t = blk * 16 + lane;
        out[1 + 2 * t]     = c0;
        out[1 + 2 * t + 1] = c1;
        pn = w * (c0 + c1);
        pd = w;
    }
    // wave-level reduction over the 16 producing lanes (others contribute 0)
    for (int m = 16; m >= 1; m >>= 1) {
        pn += __shfl_xor(pn, m, 32);
        pd += __shfl_xor(pd, m, 32);
    }
    if (lane == 0) {
        partials[2 * blk]     = pn;
        partials[2 * blk + 1] = pd;
    }
}

// Kernel 2: deterministic reduction of per-slab partials -> out[0].
__global__ __launch_bounds__(256) void tile_head_finalize(
    const float* __restrict__ partials, float* __restrict__ out)
{
    __shared__ float snum[256];
    __shared__ float sden[256];
    float n = 0.0f, d = 0.0f;
    for (int i = threadIdx.x; i < NBLK; i += 256) {
        n += partials[2 * i];
        d += partials[2 * i + 1];
    }
    snum[threadIdx.x] = n;
    sden[threadIdx.x] = d;
    __syncthreads();
    for (int s = 128; s > 0; s >>= 1) {
        if (threadIdx.x < (unsigned)s) {
            snum[threadIdx.x] += snum[threadIdx.x + s];
            sden[threadIdx.x] += sden[threadIdx.x + s];
        }
        __syncthreads();
    }
    if (threadIdx.x == 0) out[0] = snum[0] / sden[0];
}

extern "C" void kernel_launch(void* const* d_in, const int* in_sizes, int n_in,
                              void* d_out, int out_size, void* d_ws, size_t ws_size,
                              hipStream_t stream) {
    (void)in_sizes; (void)n_in; (void)out_size; (void)ws_size;
    const float* img = (const float*)d_in[0];   // images [1, NTILES, DIM]
    // d_in[1] = coords (unused by the reference computation)
    const float* Wc  = (const float*)d_in[2];   // [2, DIM]
    const float* bc  = (const float*)d_in[3];   // [2]
    const float* Wa  = (const float*)d_in[4];   // [1, DIM]
    const float* ba  = (const float*)d_in[5];   // [1]
    float* out       = (float*)d_out;           // [1 + NTILES*2]
    float* partials  = (float*)d_ws;            // [NBLK, 2]

    const int grid = (NBLK + WAVES_PER_WG - 1) / WAVES_PER_WG;  // 391
    tile_head_wmma<<<grid, 256, 0, stream>>>(img, Wc, bc, Wa, ba, out, partials);
    tile_head_finalize<<<1, 256, 0, stream>>>(partials, out);
}